// SSIM3DLossSingleChannel_59871844106334
// MI455X (gfx1250) — compile-verified
//
#include <hip/hip_runtime.h>
#include <hip/hip_bf16.h>

// SSIM3D loss, fully fused separable 11-tap Gaussian conv for MI455X (gfx1250).
// z-pass: streaming FMA from global (inputs are L2-resident: 134MB < 192MB L2).
// y-pass & x-pass: V_WMMA_F32_16X16X4_F32 band-matrix matmuls (f32 accuracy).
// SSIM map evaluated lane-locally on WMMA accumulator fragments, global f32 atomic reduce.
// Round 2: TZ 2->4 (~110KB LDS, exploits gfx1250's 320KB LDS), 256-thread blocks.

typedef __attribute__((ext_vector_type(2))) float v2f;
typedef __attribute__((ext_vector_type(8))) float v8f;

#define BATCH 8
#define DIM   128
#define TZ    4
#define TY    16
#define TX    16
#define NTHREADS 256
#define NWAVES   (NTHREADS / 32)

__device__ __forceinline__ int imin(int a, int b) { return a < b ? a : b; }

// normalized 1-D Gaussian, size 11, sigma 1.5 (3-D window is the separable product)
__device__ __forceinline__ float gw(int t) {
  constexpr float G[11] = {
    0.00102843f, 0.00759877f, 0.03600075f, 0.10936072f, 0.21300536f,
    0.26601172f, 0.21300536f, 0.10936072f, 0.03600075f, 0.00759877f,
    0.00102843f };
  return G[t];
}

// banded conv matrix G[m][k] = gw(k-m) for 0<=k-m<=10, else 0 (16x28, K-pad rows are zero)
__device__ __forceinline__ float bandv(int m, int k) {
  const int d = k - m;
  float r = 0.0f;
#pragma unroll
  for (int t = 0; t < 11; ++t) r = (d == t) ? gw(t) : r;
  return r;
}

__global__ void __launch_bounds__(NTHREADS)
ssim3d_fused(const float* __restrict__ img1, const float* __restrict__ img2,
             float* __restrict__ acc)
{
  __shared__ float S1[5][TZ][26][32];   // z-convolved, y/x halo, x padded to 32 (zeroed)
  __shared__ float S2[5][TZ][16][33];   // y-convolved, x halo, stride 33 (bank-conflict pad)
  __shared__ float red[NTHREADS];

  const int tid  = threadIdx.x;
  const int lane = tid & 31;
  const int w    = tid >> 5;

  const int x0 = blockIdx.x * TX;
  const int y0 = blockIdx.y * TY;
  const int bz = blockIdx.z;            // 8 batches * (128/TZ) z-tiles
  const int b  = bz / (DIM / TZ);
  const int z0 = (bz % (DIM / TZ)) * TZ;

  const float* base1 = img1 + b * (DIM * DIM * DIM);
  const float* base2 = img2 + b * (DIM * DIM * DIM);

  // ---------- Stage A: z-convolution of 5 channels, global -> S1 ----------
  for (int p = tid; p < 26 * 32; p += NTHREADS) {
    const int yy = p >> 5;              // 0..25  (y halo index)
    const int xx = p & 31;              // 0..31  (26..31 = zero pad for WMMA K)
    const int gy = y0 + yy - 5;
    const int gx = x0 + xx - 5;
    float m1[TZ], m2[TZ], q1[TZ], q2[TZ], cr[TZ];
#pragma unroll
    for (int zi = 0; zi < TZ; ++zi) { m1[zi]=0.f; m2[zi]=0.f; q1[zi]=0.f; q2[zi]=0.f; cr[zi]=0.f; }
    if (xx < 26 && gy >= 0 && gy < DIM && gx >= 0 && gx < DIM) {
      const int planeOff = gy * DIM + gx;
      float a[TZ + 10], c[TZ + 10];
#pragma unroll
      for (int t = 0; t < TZ + 10; ++t) {
        const int gz = z0 + t - 5;
        const bool ok = (gz >= 0) && (gz < DIM);
        const int idx = planeOff + gz * (DIM * DIM);
        a[t] = ok ? base1[idx] : 0.0f;
        c[t] = ok ? base2[idx] : 0.0f;
      }
#pragma unroll
      for (int zi = 0; zi < TZ; ++zi) {
        float s1 = 0.f, s2 = 0.f, s11 = 0.f, s22 = 0.f, s12 = 0.f;
#pragma unroll
        for (int t = 0; t < 11; ++t) {
          const float g  = gw(t);
          const float va = a[zi + t], vb = c[zi + t];
          s1 += g * va; s2 += g * vb;
          s11 += g * va * va; s22 += g * vb * vb; s12 += g * va * vb;
        }
        m1[zi] = s1; m2[zi] = s2; q1[zi] = s11; q2[zi] = s22; cr[zi] = s12;
      }
    }
#pragma unroll
    for (int zi = 0; zi < TZ; ++zi) {
      S1[0][zi][yy][xx] = m1[zi];
      S1[1][zi][yy][xx] = m2[zi];
      S1[2][zi][yy][xx] = q1[zi];
      S1[3][zi][yy][xx] = q2[zi];
      S1[4][zi][yy][xx] = cr[zi];
    }
  }

  // per-lane A-operand fragments of the constant 16x28 band matrix (shared by both WMMA stages)
  // f32 16x16x4 A layout: lanes 0-15 row M=lane, VGPR0=K0/VGPR1=K1; lanes 16-31 same rows, K2/K3
  v2f frA[7];
  {
    const int m  = lane & 15;
    const int hi = (lane & 16) ? 2 : 0;
#pragma unroll
    for (int j = 0; j < 7; ++j) {
      frA[j].x = bandv(m, 4 * j + hi);
      frA[j].y = bandv(m, 4 * j + hi + 1);
    }
  }

  __syncthreads();

  // ---------- Stage B: y-convolution via WMMA, S1 -> S2 ----------
  // tiles: 5 channels x TZ z-planes x 2 col-groups (x' cols 0..31); wave-uniform loop
  for (int ti = w; ti < 5 * TZ * 2; ti += NWAVES) {
    const int ch  = ti >> 3;                  // / (TZ*2)
    const int z   = (ti >> 1) & (TZ - 1);
    const int grp = ti & 1;
    const int n   = grp * 16 + (lane & 15);   // output column = x' index
    const int hi  = (lane & 16) ? 2 : 0;
    v8f cacc = {0.f, 0.f, 0.f, 0.f, 0.f, 0.f, 0.f, 0.f};
#pragma unroll
    for (int j = 0; j < 7; ++j) {
      const int k = 4 * j + hi;               // K = y' index (0..27, band rows >=26 are zero)
      v2f bf;
      bf.x = S1[ch][z][imin(k, 25)][n];
      bf.y = S1[ch][z][imin(k + 1, 25)][n];
      cacc = __builtin_amdgcn_wmma_f32_16x16x4_f32(
          false, frA[j], false, bf, (short)0, cacc, false, false);
    }
    const int mrow = (lane & 16) ? 8 : 0;     // D: VGPR r -> M=r (lanes0-15) / M=r+8
#pragma unroll
    for (int r = 0; r < 8; ++r)
      S2[ch][z][r + mrow][n] = cacc[r];
  }

  __syncthreads();

  // ---------- Stage C: x-convolution via WMMA + SSIM map (lane-local) ----------
  float lsum = 0.0f;
  if (w < TZ) {                               // wave-uniform: EXEC stays all-ones for WMMA
    const int z  = w;
    const int n  = lane & 15;                 // output column = y index
    const int hi = (lane & 16) ? 2 : 0;
    v8f chv[5];
#pragma unroll
    for (int ch = 0; ch < 5; ++ch) {
      v8f cacc = {0.f, 0.f, 0.f, 0.f, 0.f, 0.f, 0.f, 0.f};
#pragma unroll
      for (int j = 0; j < 7; ++j) {
        const int k = 4 * j + hi;             // K = x' index (0..27, all finite in S2)
        v2f bf;
        bf.x = S2[ch][z][n][k];
        bf.y = S2[ch][z][n][k + 1];
        cacc = __builtin_amdgcn_wmma_f32_16x16x4_f32(
            false, frA[j], false, bf, (short)0, cacc, false, false);
      }
      chv[ch] = cacc;
    }
    const float C1 = 1.0e-4f, C2 = 9.0e-4f;
#pragma unroll
    for (int i = 0; i < 8; ++i) {
      const float mu1 = chv[0][i], mu2 = chv[1][i];
      const float e11 = chv[2][i], e22 = chv[3][i], e12 = chv[4][i];
      const float mu1s = mu1 * mu1, mu2s = mu2 * mu2, mu12 = mu1 * mu2;
      const float sg1 = e11 - mu1s, sg2 = e22 - mu2s, sg12 = e12 - mu12;
      lsum += ((2.0f * mu12 + C1) * (2.0f * sg12 + C2)) /
              ((mu1s + mu2s + C1) * (sg1 + sg2 + C2));
    }
  }

  red[tid] = lsum;
  __syncthreads();
#pragma unroll
  for (int s = NTHREADS / 2; s > 0; s >>= 1) {
    if (tid < s) red[tid] += red[tid + s];
    __syncthreads();
  }
  if (tid == 0) atomicAdd(acc, red[0]);
}

__global__ void ssim3d_init(float* acc) { acc[0] = 0.0f; }

__global__ void ssim3d_final(const float* __restrict__ acc, float* __restrict__ out) {
  out[0] = 1.0f - acc[0] * (1.0f / 16777216.0f);   // mean over 8*128^3, then 1 - mean
}

extern "C" void kernel_launch(void* const* d_in, const int* in_sizes, int n_in,
                              void* d_out, int out_size, void* d_ws, size_t ws_size,
                              hipStream_t stream) {
  (void)in_sizes; (void)n_in; (void)out_size; (void)ws_size;
  const float* img1 = (const float*)d_in[0];
  const float* img2 = (const float*)d_in[1];
  float* out = (float*)d_out;
  float* acc = (float*)d_ws;                 // 4-byte accumulator in scratch

  ssim3d_init<<<1, 1, 0, stream>>>(acc);
  dim3 grid(DIM / TX, DIM / TY, BATCH * (DIM / TZ));   // 8 x 8 x 256 blocks
  ssim3d_fused<<<grid, NTHREADS, 0, stream>>>(img1, img2, acc);
  ssim3d_final<<<1, 1, 0, stream>>>(acc, out);
}